// ModelWithEdgeFeatures_76484777607334
// MI455X (gfx1250) — compile-verified
//
#include <hip/hip_runtime.h>
#include <cstdint>

#define N_NODES  10000
#define E_EDGES  320000
#define B_GRAPHS 64
#define IN_CH    128
#define EC_CH    64
#define HID      256
#define MLP_CH   512
#define NC       4
#define EPS_BN   1e-5f
#define MAX_SIZE 40.0f

typedef __attribute__((ext_vector_type(2))) float v2f;
typedef __attribute__((ext_vector_type(8))) float v8f;

// ---------------------------------------------------------------------------
// Precompute: deg_f[v] = in-degree + 1 (self loop), agg_ea[v] = sum_e ea + 1
// ---------------------------------------------------------------------------
__global__ void k_init_deg_ea(float* __restrict__ deg, float* __restrict__ agg_ea) {
    size_t tid = (size_t)blockIdx.x * blockDim.x + threadIdx.x;
    if (tid < (size_t)N_NODES * EC_CH) agg_ea[tid] = 1.0f;   // self-loop edge_attr = 1
    if (tid < N_NODES) deg[tid] = 1.0f;                      // self-loop degree
}

__global__ void k_scatter_ea(const int* __restrict__ ei, const float* __restrict__ ea,
                             float* __restrict__ deg, float* __restrict__ agg_ea) {
    size_t tid = (size_t)blockIdx.x * blockDim.x + threadIdx.x;
    int e = (int)(tid / EC_CH);
    int c = (int)(tid % EC_CH);
    if (e >= E_EDGES) return;
    int d = ei[E_EDGES + e];                                 // dst row of edge_index
    atomicAdd(&agg_ea[(size_t)d * EC_CH + c], ea[(size_t)e * EC_CH + c]);
    if (c == 0) atomicAdd(&deg[d], 1.0f);
}

// ---------------------------------------------------------------------------
// Build X[v] = [ deg_f[v]*h[v] (D) | h[v] (D, self-loop seed of agg) | agg_ea (64) ]
// ---------------------------------------------------------------------------
__global__ void k_buildX(const float* __restrict__ h, int D, const float* __restrict__ deg,
                         const float* __restrict__ agg_ea, float* __restrict__ X) {
    const int K = 2 * D + EC_CH;
    size_t tid = (size_t)blockIdx.x * blockDim.x + threadIdx.x;
    if (tid >= (size_t)N_NODES * K) return;
    int v = (int)(tid / K);
    int c = (int)(tid % K);
    float val;
    if (c < D)           val = deg[v] * h[(size_t)v * D + c];
    else if (c < 2 * D)  val = h[(size_t)v * D + (c - D)];
    else                 val = agg_ea[(size_t)v * EC_CH + (c - 2 * D)];
    X[tid] = val;   // tid == v*K + c (row-major, stride K)
}

// Scatter-add h[src] into the agg slot of X[dst]
__global__ void k_scatterX(const int* __restrict__ ei, const float* __restrict__ h,
                           int D, float* __restrict__ X) {
    const int K = 2 * D + EC_CH;
    size_t tid = (size_t)blockIdx.x * blockDim.x + threadIdx.x;
    int e = (int)(tid / D);
    int c = (int)(tid % D);
    if (e >= E_EDGES) return;
    int s = ei[e];
    int d = ei[E_EDGES + e];
    atomicAdd(&X[(size_t)d * K + D + c], h[(size_t)s * D + c]);
}

// ---------------------------------------------------------------------------
// Pack W (K x 256 row-major) into K-paired layout so the per-lane B fragment
// {W[k][col], W[k+1][col]} is one contiguous 8-byte load:
//   Wp[(k/2)*HID + col] = { W[k][col], W[k+1][col] }   (stored as float2)
// ---------------------------------------------------------------------------
__global__ void k_packW(const float* __restrict__ W, int K, float* __restrict__ Wp) {
    size_t tid = (size_t)blockIdx.x * blockDim.x + threadIdx.x;
    if (tid >= (size_t)(K / 2) * HID) return;
    size_t p   = tid / HID;
    size_t col = tid % HID;
    Wp[tid * 2]     = W[(2 * p) * HID + col];
    Wp[tid * 2 + 1] = W[(2 * p + 1) * HID + col];
}

// ---------------------------------------------------------------------------
// out[v] = relu( X[v] @ W + deg_f[v]*bias )   via V_WMMA_F32_16X16X4_F32
//   block = 128 threads = 4 waves; block covers 16 rows x 256 cols; wave w
//   owns columns [64w, 64w+64) with four 16x16 f32 accumulators.
//   Per K-step: 1 x b64 A load + 4 x b64 B loads feed 4 WMMAs.
//   N = 625*16 exactly -> no bounds checks, EXEC stays all-1s.
// ---------------------------------------------------------------------------
__global__ void __launch_bounds__(128)
k_gemm_relu(const float* __restrict__ X, int K,
            const float* __restrict__ Wp, const float* __restrict__ bias,
            const float* __restrict__ deg, float* __restrict__ out) {
    const int lane = threadIdx.x & 31;
    const int wave = threadIdx.x >> 5;      // 0..3
    const int row0 = blockIdx.x * 16;
    const int col0 = wave * 64;
    const int m    = lane & 15;             // A row / C column within tile
    const int half = lane >> 4;             // which K pair this lane holds
    const int n    = m;

    const float* Xrow = X + (size_t)(row0 + m) * K + 2 * half;

    v8f acc0 = {}, acc1 = {}, acc2 = {}, acc3 = {};
    for (int k0 = 0; k0 < K; k0 += 4) {
        v2f a = *(const v2f*)(Xrow + k0);
        // pair index for this lane: (k0 + 2*half)/2 = k0/2 + half
        const float* Wpk = Wp + (size_t)(k0 / 2 + half) * (HID * 2);
        v2f b0 = *(const v2f*)(Wpk + 2 * (col0 + n));
        v2f b1 = *(const v2f*)(Wpk + 2 * (col0 + 16 + n));
        v2f b2 = *(const v2f*)(Wpk + 2 * (col0 + 32 + n));
        v2f b3 = *(const v2f*)(Wpk + 2 * (col0 + 48 + n));
        acc0 = __builtin_amdgcn_wmma_f32_16x16x4_f32(false, a, false, b0,
                                                     (short)0, acc0, false, false);
        acc1 = __builtin_amdgcn_wmma_f32_16x16x4_f32(false, a, false, b1,
                                                     (short)0, acc1, false, false);
        acc2 = __builtin_amdgcn_wmma_f32_16x16x4_f32(false, a, false, b2,
                                                     (short)0, acc2, false, false);
        acc3 = __builtin_amdgcn_wmma_f32_16x16x4_f32(false, a, false, b3,
                                                     (short)0, acc3, false, false);
    }

#pragma unroll
    for (int i = 0; i < 8; ++i) {
        int row = row0 + i + 8 * half;
        float dv = deg[row];
        float* orow = out + (size_t)row * HID + col0 + n;
        float o0 = acc0[i] + dv * bias[col0 + n];
        float o1 = acc1[i] + dv * bias[col0 + 16 + n];
        float o2 = acc2[i] + dv * bias[col0 + 32 + n];
        float o3 = acc3[i] + dv * bias[col0 + 48 + n];
        orow[0]  = o0 > 0.0f ? o0 : 0.0f;
        orow[16] = o1 > 0.0f ? o1 : 0.0f;
        orow[32] = o2 > 0.0f ? o2 : 0.0f;
        orow[48] = o3 > 0.0f ? o3 : 0.0f;
    }
}

// ---------------------------------------------------------------------------
// BatchNorm (training stats) + relu
// ---------------------------------------------------------------------------
__global__ void k_zero(float* __restrict__ p, int n) {
    int tid = blockIdx.x * blockDim.x + threadIdx.x;
    if (tid < n) p[tid] = 0.0f;
}

__global__ void k_stats(const float* __restrict__ h1, float* __restrict__ sum,
                        float* __restrict__ sumsq) {
    const int c  = threadIdx.x;          // 256 channels
    const int r0 = blockIdx.x * 40;      // 250 blocks * 40 rows = 10000
    float s = 0.0f, ss = 0.0f;
    for (int r = r0; r < r0 + 40; ++r) {
        float v = h1[(size_t)r * HID + c];
        s  += v;
        ss += v * v;
    }
    atomicAdd(&sum[c], s);
    atomicAdd(&sumsq[c], ss);
}

__global__ void k_bnrelu(const float* __restrict__ h1, const float* __restrict__ sum,
                         const float* __restrict__ sumsq, const float* __restrict__ g,
                         const float* __restrict__ be, float* __restrict__ hout) {
    size_t tid = (size_t)blockIdx.x * blockDim.x + threadIdx.x;
    if (tid >= (size_t)N_NODES * HID) return;
    int c = (int)(tid % HID);
    float inv_n = 1.0f / (float)N_NODES;
    float mu  = sum[c] * inv_n;
    float var = sumsq[c] * inv_n - mu * mu;
    float rs  = rsqrtf(var + EPS_BN);
    float x   = (h1[tid] - mu) * rs * g[c] + be[c];
    hout[tid] = x > 0.0f ? x : 0.0f;
}

// ---------------------------------------------------------------------------
// Pooling + MLP head
// ---------------------------------------------------------------------------
__global__ void k_pool(const float* __restrict__ h, const int* __restrict__ batch,
                       float* __restrict__ pooled, float* __restrict__ counts) {
    size_t tid = (size_t)blockIdx.x * blockDim.x + threadIdx.x;
    if (tid >= (size_t)N_NODES * HID) return;
    int v = (int)(tid / HID);
    int c = (int)(tid % HID);
    int b = batch[v];
    atomicAdd(&pooled[(size_t)b * HID + c], h[tid]);
    if (c == 0) atomicAdd(&counts[b], 1.0f);
}

__global__ void k_fc1(const float* __restrict__ pooled, const float* __restrict__ counts,
                      const float* __restrict__ neighbor, const float* __restrict__ w,
                      const float* __restrict__ b, float* __restrict__ out) {
    int tid = blockIdx.x * blockDim.x + threadIdx.x;     // 64*512
    if (tid >= B_GRAPHS * MLP_CH) return;
    int bg = tid / MLP_CH;
    int j  = tid % MLP_CH;
    float acc = b[j];
    for (int k = 0; k < HID; ++k)
        acc += pooled[(size_t)bg * HID + k] * w[(size_t)k * MLP_CH + j];
    acc += (counts[bg] * (1.0f / MAX_SIZE)) * w[(size_t)HID * MLP_CH + j];
    for (int k = 0; k < IN_CH; ++k)
        acc += neighbor[(size_t)bg * IN_CH + k] * w[(size_t)(HID + 1 + k) * MLP_CH + j];
    out[tid] = acc > 0.0f ? acc : 0.0f;
}

__global__ void k_fc2(const float* __restrict__ z1, const float* __restrict__ w,
                      const float* __restrict__ b, float* __restrict__ out) {
    int tid = blockIdx.x * blockDim.x + threadIdx.x;     // 64*4
    if (tid >= B_GRAPHS * NC) return;
    int bg = tid / NC;
    int n  = tid % NC;
    float acc = b[n];
    for (int j = 0; j < MLP_CH; ++j)
        acc += z1[(size_t)bg * MLP_CH + j] * w[(size_t)j * NC + n];
    out[tid] = acc;
}

// ---------------------------------------------------------------------------
extern "C" void kernel_launch(void* const* d_in, const int* in_sizes, int n_in,
                              void* d_out, int out_size, void* d_ws, size_t ws_size,
                              hipStream_t stream) {
    const float* x        = (const float*)d_in[0];
    const int*   ei       = (const int*)  d_in[1];
    const float* ea       = (const float*)d_in[2];
    const int*   batch    = (const int*)  d_in[3];
    const float* neighbor = (const float*)d_in[4];
    const float* Ws[3]    = {(const float*)d_in[5],  (const float*)d_in[9],  (const float*)d_in[13]};
    const float* bs[3]    = {(const float*)d_in[6],  (const float*)d_in[10], (const float*)d_in[14]};
    const float* gs[3]    = {(const float*)d_in[7],  (const float*)d_in[11], (const float*)d_in[15]};
    const float* bes[3]   = {(const float*)d_in[8],  (const float*)d_in[12], (const float*)d_in[16]};
    const float* fc1_w    = (const float*)d_in[17];
    const float* fc1_b    = (const float*)d_in[18];
    const float* fc2_w    = (const float*)d_in[19];
    const float* fc2_b    = (const float*)d_in[20];
    float* out = (float*)d_out;

    // workspace layout (floats; all offsets even -> 8B alignment for b64 loads)
    float* ws = (float*)d_ws;
    size_t off = 0;
    float* deg    = ws + off; off += 10240;                      // N (padded)
    float* agg_ea = ws + off; off += (size_t)N_NODES * EC_CH;    // 640000
    float* X      = ws + off; off += (size_t)N_NODES * 576;      // worst-case K
    float* h1     = ws + off; off += (size_t)N_NODES * HID;
    float* hA     = ws + off; off += (size_t)N_NODES * HID;
    float* hB     = ws + off; off += (size_t)N_NODES * HID;
    float* Wp     = ws + off; off += (size_t)576 * HID;          // packed weights
    float* bsum   = ws + off; off += 256;
    float* bssq   = ws + off; off += 256;
    float* pooled = ws + off; off += (size_t)B_GRAPHS * HID;
    float* counts = ws + off; off += 64;
    float* z1     = ws + off; off += (size_t)B_GRAPHS * MLP_CH;

    const int T = 256;
    auto blocks = [](size_t n, int t) { return (unsigned)((n + t - 1) / t); };

    // edge-only precompute (shared by all 3 layers)
    k_init_deg_ea<<<blocks((size_t)N_NODES * EC_CH, T), T, 0, stream>>>(deg, agg_ea);
    k_scatter_ea<<<blocks((size_t)E_EDGES * EC_CH, T), T, 0, stream>>>(ei, ea, deg, agg_ea);

    // three message-passing layers
    const float* hin[3]  = {x, hA, hB};
    float*       hout[3] = {hA, hB, hA};
    for (int l = 0; l < 3; ++l) {
        const int D = (l == 0) ? IN_CH : HID;
        const int K = 2 * D + EC_CH;                 // 320 or 576
        k_buildX<<<blocks((size_t)N_NODES * K, T), T, 0, stream>>>(hin[l], D, deg, agg_ea, X);
        k_scatterX<<<blocks((size_t)E_EDGES * D, T), T, 0, stream>>>(ei, hin[l], D, X);
        k_packW<<<blocks((size_t)(K / 2) * HID, T), T, 0, stream>>>(Ws[l], K, Wp);
        k_gemm_relu<<<N_NODES / 16, 128, 0, stream>>>(X, K, Wp, bs[l], deg, h1);
        k_zero<<<2, 256, 0, stream>>>(bsum, 512);    // bsum + bssq are contiguous
        k_stats<<<N_NODES / 40, 256, 0, stream>>>(h1, bsum, bssq);
        k_bnrelu<<<blocks((size_t)N_NODES * HID, T), T, 0, stream>>>(h1, bsum, bssq,
                                                                     gs[l], bes[l], hout[l]);
    }

    // pooling + head
    k_zero<<<blocks((size_t)B_GRAPHS * HID + 64, T), T, 0, stream>>>(pooled, B_GRAPHS * HID + 64);
    k_pool<<<blocks((size_t)N_NODES * HID, T), T, 0, stream>>>(hA, batch, pooled, counts);
    k_fc1<<<blocks((size_t)B_GRAPHS * MLP_CH, T), T, 0, stream>>>(pooled, counts, neighbor,
                                                                  fc1_w, fc1_b, z1);
    k_fc2<<<1, 256, 0, stream>>>(z1, fc2_w, fc2_b, out);
}